// GroupedSubMConv3d_83640193122771
// MI455X (gfx1250) — compile-verified
//
#include <hip/hip_runtime.h>

typedef float v2f __attribute__((ext_vector_type(2)));
typedef float v4f __attribute__((ext_vector_type(4)));
typedef float v8f __attribute__((ext_vector_type(8)));

#define N_VOX   400000
#define KVOL    27
#define GROUPS  4
#define C_TOT   64
#define TILE    16
#define WAVES   8
#define BLOCK   256
#define TGROUPS (N_VOX / TILE / WAVES)          // 3125 tile-groups of 8 tiles
#define GRID    512                             // persistent blocks, 2 per WGP
#define W_ELEMS (KVOL * GROUPS * 256)           // 27648 floats = 110592 B LDS

// Zero row used as the gather target for inactive neighbor sites (idx < 0).
__global__ void zrow_init_kernel(float* __restrict__ z) {
    z[threadIdx.x] = 0.0f;
}

__global__ __launch_bounds__(BLOCK)
void subm_conv_wmma_f32(const float* __restrict__ feat,
                        const float* __restrict__ weight,
                        const float* __restrict__ bias,
                        const int*   __restrict__ nb,
                        const float* __restrict__ zrow,
                        float*       __restrict__ out)
{
    // ci permutation: WMMA step j consumes ci = {2j, 2j+1, 2j+8, 2j+9}
    // (K-slot 0,1 -> ci 2j,2j+1 ; K-slot 2,3 -> ci 2j+8,2j+9).
    // LDS B layout: WL[k][g][kh][n][j][p] = W[g][k][ci = 2j + 8*kh + p][co = n]
    // -> per (k,g,kh,n): the 8 floats for j=0..3 are contiguous (two b128 loads).
    extern __shared__ float wl[];

    const int tid = threadIdx.x;
    {
        // thread = (ci, co); swizzle indices are loop-invariant.
        const int co = tid & 15, ci = tid >> 4;
        const int kh = ci >> 3, j = (ci >> 1) & 3, p = ci & 1;
        const int lofs = kh * 128 + co * 8 + j * 2 + p;
        const float* wg = weight + tid;               // coalesced b32 reads
        for (int g = 0; g < GROUPS; ++g)
            for (int k = 0; k < KVOL; ++k)
                wl[k * 1024 + g * 256 + lofs] = wg[(size_t)(g * KVOL + k) * 256];
    }
    __syncthreads();

    const int lane = tid & 31;
    const int wave = tid >> 5;
    const int m    = lane & 15;   // A row (voxel in tile) == B/C column n
    const int kh   = lane >> 4;   // K-pair selector (lanes 16-31 hold K=2,3)
    const int n    = m;

    float bv[GROUPS];
    #pragma unroll
    for (int g = 0; g < GROUPS; ++g) bv[g] = bias[g * 16 + n];

    const float* wl_lane = wl + kh * 128 + n * 8;

    for (int tg = blockIdx.x; tg < TGROUPS; tg += gridDim.x) {
        const int tile = tg * WAVES + wave;
        const int* __restrict__ nbrow = nb + (size_t)(tile * TILE + m) * KVOL;

        // Hoist all 27 neighbor indices (contiguous per lane) into registers:
        // one clause of loads, one wait, no per-k dependent index fetch.
        int idxs[KVOL];
        #pragma unroll
        for (int k = 0; k < KVOL; ++k)
            idxs[k] = __builtin_nontemporal_load(nbrow + k);

        v8f zero = {};
        v8f acc[GROUPS];
        #pragma unroll
        for (int g = 0; g < GROUPS; ++g) acc[g] = zero;

        #pragma unroll 3
        for (int k = 0; k < KVOL; ++k) {
            const int idx = idxs[k];
            // Address-select instead of data masking: inactive sites gather zeros.
            const float* frow = (idx >= 0) ? (feat + (size_t)idx * C_TOT) : zrow;
            const float* ap = frow + kh * 8;      // + imm offsets g*64 + j2*16 B
            const float* bp = wl_lane + k * 1024; // + imm offsets g*1024 + j2*16 B
            #pragma unroll
            for (int g = 0; g < GROUPS; ++g) {
                #pragma unroll
                for (int j2 = 0; j2 < 2; ++j2) {
                    // One b128 of A and one b128 of B feed two WMMA steps.
                    v4f a4 = *(const v4f*)(ap + g * 16 + j2 * 4);
                    v4f b4 = *(const v4f*)(bp + g * 256 + j2 * 4);
                    v2f a0 = __builtin_shufflevector(a4, a4, 0, 1);
                    v2f a1 = __builtin_shufflevector(a4, a4, 2, 3);
                    v2f b0 = __builtin_shufflevector(b4, b4, 0, 1);
                    v2f b1 = __builtin_shufflevector(b4, b4, 2, 3);
                    acc[g] = __builtin_amdgcn_wmma_f32_16x16x4_f32(
                        false, a0, false, b0, (short)0, acc[g], false, false);
                    acc[g] = __builtin_amdgcn_wmma_f32_16x16x4_f32(
                        false, a1, false, b1, (short)0, acc[g], false, false);
                }
            }
        }

        // D layout: VGPR r holds rows {r, r+8}; NT stores keep feat hot in L2.
        #pragma unroll
        for (int g = 0; g < GROUPS; ++g) {
            #pragma unroll
            for (int r = 0; r < 8; ++r) {
                const int row = tile * TILE + r + 8 * kh;
                __builtin_nontemporal_store(acc[g][r] + bv[g],
                                            out + (size_t)row * C_TOT + g * 16 + n);
            }
        }
    }
}

extern "C" void kernel_launch(void* const* d_in, const int* in_sizes, int n_in,
                              void* d_out, int out_size, void* d_ws, size_t ws_size,
                              hipStream_t stream) {
    (void)in_sizes; (void)n_in; (void)ws_size; (void)out_size;
    const float* feat   = (const float*)d_in[0];
    const float* weight = (const float*)d_in[1];
    const float* bias   = (const float*)d_in[2];
    const int*   nb     = (const int*)d_in[3];
    float*       zrow   = (float*)d_ws;
    float*       out    = (float*)d_out;

    zrow_init_kernel<<<1, C_TOT, 0, stream>>>(zrow);

    const size_t lds = W_ELEMS * sizeof(float);   // 110592 B
    subm_conv_wmma_f32<<<GRID, BLOCK, lds, stream>>>(feat, weight, bias, nb,
                                                     zrow, out);
}